// LoRIA3DLUT_2448131359066
// MI455X (gfx1250) — compile-verified
//
#include <hip/hip_runtime.h>
#include <hip/hip_bf16.h>

#define GG   33
#define KK   8
#define RR   8
#define BB   4
#define HL   256
#define WL   256
#define HF   1080
#define WF   1920
#define G3   (GG*GG*GG)        /* 35937  */
#define NC   (GG*GG*3)         /* 3267 columns per x-row of the bxyzc grid */
#define DSZ  (BB*G3*3)         /* 431244 delta/L elements */
#define OIMG (BB*3*HF*WF)      /* 24883200 */

// output offsets (flat, in floats): out | alpha | delta | L | delta_norm
#define ALPHA_OFF  (OIMG)
#define DELTA_OFF  (ALPHA_OFF + BB*KK)
#define L_OFF      (DELTA_OFF + DSZ)
#define NORM_OFF   (L_OFF + DSZ)

#define NORM_BLOCKS 128

typedef __attribute__((ext_vector_type(2))) float v2f;
typedef __attribute__((ext_vector_type(4))) float v4f;
typedef __attribute__((ext_vector_type(8))) float v8f;

// ---------------- conv1: 3->16, 3x3, stride 2, pad 1, ReLU -----------------
__global__ void conv1_relu(const float* __restrict__ img,   // (B,3,256,256)
                           const float* __restrict__ wgt,   // (16,3,3,3)
                           const float* __restrict__ bias,  // (16)
                           float* __restrict__ out)         // (B,16,128,128)
{
    int idx = blockIdx.x * blockDim.x + threadIdx.x;
    if (idx >= BB*16*128*128) return;
    int ow = idx & 127;
    int oh = (idx >> 7) & 127;
    int oc = (idx >> 14) & 15;
    int b  = idx >> 18;
    float acc = bias[oc];
    for (int ic = 0; ic < 3; ++ic) {
        const float* ip = img + ((long)(b*3 + ic)) * HL * WL;
        const float* wp = wgt + ((oc*3 + ic)*9);
        #pragma unroll
        for (int kh = 0; kh < 3; ++kh) {
            int ih = 2*oh - 1 + kh;
            if ((unsigned)ih >= (unsigned)HL) continue;
            #pragma unroll
            for (int kw = 0; kw < 3; ++kw) {
                int iw = 2*ow - 1 + kw;
                if ((unsigned)iw >= (unsigned)WL) continue;
                acc += ip[ih*WL + iw] * wp[kh*3 + kw];
            }
        }
    }
    out[idx] = fmaxf(acc, 0.0f);
}

// ------------- conv2: 16->32, 3x3, stride 2, pad 1, ReLU + mean ------------
__global__ void conv2_relu_pool(const float* __restrict__ h1,   // (B,16,128,128)
                                const float* __restrict__ wgt,  // (32,16,3,3)
                                const float* __restrict__ bias, // (32)
                                float* __restrict__ feat)       // (B,32)
{
    int oc = blockIdx.x;   // 32
    int b  = blockIdx.y;   // 4
    int tid = threadIdx.x; // 256
    float sum = 0.0f;
    for (int p = tid; p < 64*64; p += 256) {
        int ow = p & 63, oh = p >> 6;
        float acc = bias[oc];
        for (int ic = 0; ic < 16; ++ic) {
            const float* ip = h1 + ((long)(b*16 + ic)) * 128 * 128;
            const float* wp = wgt + ((oc*16 + ic)*9);
            #pragma unroll
            for (int kh = 0; kh < 3; ++kh) {
                int ih = 2*oh - 1 + kh;
                if ((unsigned)ih >= 128u) continue;
                #pragma unroll
                for (int kw = 0; kw < 3; ++kw) {
                    int iw = 2*ow - 1 + kw;
                    if ((unsigned)iw >= 128u) continue;
                    acc += ip[ih*128 + iw] * wp[kh*3 + kw];
                }
            }
        }
        sum += fmaxf(acc, 0.0f);
    }
    __shared__ float red[256];
    red[tid] = sum;
    __syncthreads();
    for (int s = 128; s > 0; s >>= 1) {
        if (tid < s) red[tid] += red[tid + s];
        __syncthreads();
    }
    if (tid == 0) feat[b*32 + oc] = red[0] * (1.0f / 4096.0f);
}

// ------------------------------- FC heads ----------------------------------
__global__ void heads(const float* __restrict__ hw, const float* __restrict__ hr,
                      const float* __restrict__ wpfc_w, const float* __restrict__ wpfc_b,
                      const float* __restrict__ fcu_w, const float* __restrict__ fcu_b,
                      const float* __restrict__ fcv_w, const float* __restrict__ fcv_b,
                      const float* __restrict__ fcw_w, const float* __restrict__ fcw_b,
                      const float* __restrict__ fcc_w, const float* __restrict__ fcc_b,
                      float* __restrict__ alpha,   // (B,K) -> d_out
                      float* __restrict__ u, float* __restrict__ v,
                      float* __restrict__ w, float* __restrict__ c)
{
    int t = blockIdx.x * blockDim.x + threadIdx.x;
    if (t < BB*KK) {
        int b = t / KK, k = t % KK;
        float a = wpfc_b[k];
        for (int j = 0; j < 32; ++j) a += hw[b*32 + j] * wpfc_w[k*32 + j];
        alpha[t] = a;
    }
    if (t < BB*RR*GG) {                 // 1056: u,v,w heads
        int b = t / (RR*GG), i = t % (RR*GG);
        float su = fcu_b[i], sv = fcv_b[i], sw = fcw_b[i];
        for (int j = 0; j < 32; ++j) {
            float h = hr[b*32 + j];
            su += h * fcu_w[i*32 + j];
            sv += h * fcv_w[i*32 + j];
            sw += h * fcw_w[i*32 + j];
        }
        u[t] = su; v[t] = sv; w[t] = sw;
    }
    if (t < BB*RR*3) {                  // 96: c head
        int b = t / (RR*3), i = t % (RR*3);
        float sc = fcc_b[i];
        for (int j = 0; j < 32; ++j) sc += hr[b*32 + j] * fcc_w[i*32 + j];
        c[t] = sc;
    }
}

// ---- delta = rank-R CP expansion as GEMM (33x8)@(8x3267) via f32 WMMA -----
// One wave per 16x16 output tile.  D[m][n] = sum_r u[r][x0+m] * P[r][n0+n],
// P[r][n] = v[r][y]*w[r][z]*c[r][ch] with n = (y*33+z)*3+ch.
// Also fuses L = alpha@bases + delta (K=8 contraction per element).
__global__ void delta_wmma(const float* __restrict__ u, const float* __restrict__ v,
                           const float* __restrict__ w, const float* __restrict__ c,
                           const float* __restrict__ alpha,  // (B,K)
                           const float* __restrict__ bases,  // (K, G3*3)
                           float* __restrict__ delta,        // (B, G3*3)
                           float* __restrict__ Lout)         // (B, G3*3)
{
    int tileN = blockIdx.x;          // 0..204
    int tileM = blockIdx.y;          // 0..2
    int b     = blockIdx.z;          // 0..3
    int lane  = threadIdx.x;         // 0..31 (wave32)
    int half  = lane >> 4;
    int l     = lane & 15;
    int x0 = tileM * 16;
    int n0 = tileN * 16;
    int myx = x0 + l;                // A row (x index), may exceed 32
    int myn = n0 + l;                // B col, may exceed 3266
    bool arow_ok = (myx < GG);
    bool ncol_ok = (myn < NC);
    // decode n -> (y,z,ch), clamp for safe loads (zeroed below when OOB)
    int ch = myn % 3;
    int tzy = myn / 3;
    int z   = tzy % GG;
    int y   = tzy / GG; if (y > GG-1) y = GG-1;
    int xi  = arow_ok ? myx : (GG-1);

    v8f acc = {};
    #pragma unroll
    for (int r0 = 0; r0 < RR; r0 += 4) {
        int rA0 = r0 + 2*half;       // this lane's two K slices
        int rA1 = rA0 + 1;
        // A-operand: A[l][2*half + j] = u[b][r0+2*half+j][x]
        float a0 = u[(b*RR + rA0)*GG + xi];
        float a1 = u[(b*RR + rA1)*GG + xi];
        // B-operand: B[2*half + j][l] = v*w*c
        float b0 = v[(b*RR + rA0)*GG + y] * w[(b*RR + rA0)*GG + z] * c[(b*RR + rA0)*3 + ch];
        float b1 = v[(b*RR + rA1)*GG + y] * w[(b*RR + rA1)*GG + z] * c[(b*RR + rA1)*3 + ch];
        v2f av, bv;
        av.x = arow_ok ? a0 : 0.0f;
        av.y = arow_ok ? a1 : 0.0f;
        bv.x = ncol_ok ? b0 : 0.0f;
        bv.y = ncol_ok ? b1 : 0.0f;
        acc = __builtin_amdgcn_wmma_f32_16x16x4_f32(
                  false, av, false, bv, (short)0, acc, false, false);
    }

    // D layout: VGPR j, lane (half,l) -> row x0 + j + 8*half, col myn
    #pragma unroll
    for (int j = 0; j < 8; ++j) {
        int xr = x0 + j + 8*half;
        if (xr < GG && ncol_ok) {
            float d = acc[j];
            long off = (long)b * (G3*3) + (long)xr * NC + myn;
            delta[off] = d;
            float Lv = d;
            #pragma unroll
            for (int k = 0; k < KK; ++k)
                Lv += alpha[b*KK + k] * bases[(long)k * (G3*3) + (long)xr * NC + myn];
            Lout[off] = Lv;
        }
    }
}

// -------- deterministic mean(|delta|): two-stage fixed-order reduction -----
__global__ void delta_norm_partial(const float* __restrict__ delta,
                                   float* __restrict__ partial)  // (NORM_BLOCKS)
{
    __shared__ float red[256];
    int blk = blockIdx.x;
    int tid = threadIdx.x;
    const int per = (DSZ + NORM_BLOCKS - 1) / NORM_BLOCKS;   // contiguous chunk
    int lo = blk * per;
    int hi = lo + per; if (hi > DSZ) hi = DSZ;
    float s = 0.0f;
    for (int i = lo + tid; i < hi; i += 256) s += fabsf(delta[i]);
    red[tid] = s;
    __syncthreads();
    for (int k = 128; k > 0; k >>= 1) {
        if (tid < k) red[tid] += red[tid + k];
        __syncthreads();
    }
    if (tid == 0) partial[blk] = red[0];
}

__global__ void delta_norm_final(const float* __restrict__ partial,
                                 float* __restrict__ outp)
{
    __shared__ float red[NORM_BLOCKS];
    int tid = threadIdx.x;           // NORM_BLOCKS threads
    red[tid] = partial[tid];
    __syncthreads();
    for (int k = NORM_BLOCKS/2; k > 0; k >>= 1) {
        if (tid < k) red[tid] += red[tid + k];
        __syncthreads();
    }
    if (tid == 0) outp[0] = red[0] / (float)DSZ;
}

// -------- trilinear LUT apply: the only HBM-bound kernel (~199 MB) ---------
// 4 pixels per thread: b128 non-temporal streaming (plane loads/stores bypass
// L2 retention so the 1.7 MB LUT stays resident); gathers use default RT.
__global__ void trilinear_apply(const float* __restrict__ img,  // (B,3,HF,WF)
                                const float* __restrict__ L,    // (B, G3*3)
                                float* __restrict__ out)        // (B,3,HF,WF)
{
    const long plane  = (long)HF * WF;        // 2073600 (multiple of 4)
    const long qplane = plane >> 2;           // 518400 float4's per plane
    long q = (long)blockIdx.x * blockDim.x + threadIdx.x;   // quad index
    const long NQ = (long)BB * qplane;
    if (q >= NQ) return;
    int  b  = (int)(q / qplane);
    long qh = q - (long)b * qplane;

    const v4f* ip = (const v4f*)(img + (long)b * 3 * plane) + qh;
    v4f r4 = __builtin_nontemporal_load(ip);
    v4f g4 = __builtin_nontemporal_load(ip + qplane);
    v4f b4 = __builtin_nontemporal_load(ip + 2*qplane);

    const float scale = (float)(GG - 1);
    const float hicl  = scale - 1e-6f;
    const float* Lb = L + (long)b * (G3*3);

    v4f o0, o1, o2;
    #pragma unroll
    for (int i = 0; i < 4; ++i) {
        float fx = fminf(fmaxf(r4[i] * scale, 0.0f), hicl);
        float fy = fminf(fmaxf(g4[i] * scale, 0.0f), hicl);
        float fz = fminf(fmaxf(b4[i] * scale, 0.0f), hicl);
        int x0i = (int)floorf(fx); if (x0i > GG-1) x0i = GG-1;
        int y0i = (int)floorf(fy); if (y0i > GG-1) y0i = GG-1;
        int z0i = (int)floorf(fz); if (z0i > GG-1) z0i = GG-1;
        int x1i = min(x0i + 1, GG-1);
        int y1i = min(y0i + 1, GG-1);
        int z1i = min(z0i + 1, GG-1);
        float xd = fx - (float)x0i;
        float yd = fy - (float)y0i;
        float zd = fz - (float)z0i;

        int i000 = ((x0i*GG + y0i)*GG + z0i)*3;
        int i100 = ((x1i*GG + y0i)*GG + z0i)*3;
        int i010 = ((x0i*GG + y1i)*GG + z0i)*3;
        int i110 = ((x1i*GG + y1i)*GG + z0i)*3;
        int i001 = ((x0i*GG + y0i)*GG + z1i)*3;
        int i101 = ((x1i*GG + y0i)*GG + z1i)*3;
        int i011 = ((x0i*GG + y1i)*GG + z1i)*3;
        int i111 = ((x1i*GG + y1i)*GG + z1i)*3;

        float w000 = (1.f-xd)*(1.f-yd)*(1.f-zd);
        float w100 = xd      *(1.f-yd)*(1.f-zd);
        float w010 = (1.f-xd)*yd      *(1.f-zd);
        float w110 = xd      *yd      *(1.f-zd);
        float w001 = (1.f-xd)*(1.f-yd)*zd;
        float w101 = xd      *(1.f-yd)*zd;
        float w011 = (1.f-xd)*yd      *zd;
        float w111 = xd      *yd      *zd;

        #pragma unroll
        for (int cch = 0; cch < 3; ++cch) {
            float val = Lb[i000+cch]*w000 + Lb[i100+cch]*w100
                      + Lb[i010+cch]*w010 + Lb[i110+cch]*w110
                      + Lb[i001+cch]*w001 + Lb[i101+cch]*w101
                      + Lb[i011+cch]*w011 + Lb[i111+cch]*w111;
            if (cch == 0) o0[i] = val;
            else if (cch == 1) o1[i] = val;
            else o2[i] = val;
        }
    }

    v4f* op = (v4f*)(out + (long)b * 3 * plane) + qh;
    __builtin_nontemporal_store(o0, op);
    __builtin_nontemporal_store(o1, op + qplane);
    __builtin_nontemporal_store(o2, op + 2*qplane);
}

extern "C" void kernel_launch(void* const* d_in, const int* in_sizes, int n_in,
                              void* d_out, int out_size, void* d_ws, size_t ws_size,
                              hipStream_t stream)
{
    const float* img_lr   = (const float*)d_in[0];
    const float* img_full = (const float*)d_in[1];
    const float* bases    = (const float*)d_in[2];
    const float* wp_c1_w  = (const float*)d_in[3];
    const float* wp_c1_b  = (const float*)d_in[4];
    const float* wp_c2_w  = (const float*)d_in[5];
    const float* wp_c2_b  = (const float*)d_in[6];
    const float* wp_fc_w  = (const float*)d_in[7];
    const float* wp_fc_b  = (const float*)d_in[8];
    const float* rp_c1_w  = (const float*)d_in[9];
    const float* rp_c1_b  = (const float*)d_in[10];
    const float* rp_c2_w  = (const float*)d_in[11];
    const float* rp_c2_b  = (const float*)d_in[12];
    const float* fcu_w    = (const float*)d_in[13];
    const float* fcu_b    = (const float*)d_in[14];
    const float* fcv_w    = (const float*)d_in[15];
    const float* fcv_b    = (const float*)d_in[16];
    const float* fcw_w    = (const float*)d_in[17];
    const float* fcw_b    = (const float*)d_in[18];
    const float* fcc_w    = (const float*)d_in[19];
    const float* fcc_b    = (const float*)d_in[20];

    float* out   = (float*)d_out;
    float* alpha = out + ALPHA_OFF;
    float* delta = out + DELTA_OFF;
    float* Lmat  = out + L_OFF;
    float* dnorm = out + NORM_OFF;

    float* ws = (float*)d_ws;
    float* h1 = ws;                       // (4,16,128,128) = 1048576 floats, reused per encoder
    float* hw = ws + 1048576;             // (4,32)
    float* hr = hw + 128;                 // (4,32)
    float* ub = hr + 128;                 // (4,8,33) = 1056
    float* vb = ub + 1056;
    float* wb = vb + 1056;
    float* cb = wb + 1056;                // (4,8,3) = 96
    float* np = cb + 96;                  // (NORM_BLOCKS) norm partials

    // weight-predictor encoder
    conv1_relu<<<4096, 256, 0, stream>>>(img_lr, wp_c1_w, wp_c1_b, h1);
    conv2_relu_pool<<<dim3(32, 4), 256, 0, stream>>>(h1, wp_c2_w, wp_c2_b, hw);
    // residual-predictor encoder (reuses h1 scratch)
    conv1_relu<<<4096, 256, 0, stream>>>(img_lr, rp_c1_w, rp_c1_b, h1);
    conv2_relu_pool<<<dim3(32, 4), 256, 0, stream>>>(h1, rp_c2_w, rp_c2_b, hr);
    // FC heads -> alpha (to d_out) and u/v/w/c factors (to ws)
    heads<<<5, 256, 0, stream>>>(hw, hr, wp_fc_w, wp_fc_b,
                                 fcu_w, fcu_b, fcv_w, fcv_b, fcw_w, fcw_b,
                                 fcc_w, fcc_b, alpha, ub, vb, wb, cb);
    // delta + L via f32 WMMA (3 M-tiles x 205 N-tiles x 4 batches, 1 wave each)
    delta_wmma<<<dim3(205, 3, 4), 32, 0, stream>>>(ub, vb, wb, cb, alpha, bases,
                                                   delta, Lmat);
    // deterministic mean(|delta|), two-stage fixed-order reduction
    delta_norm_partial<<<NORM_BLOCKS, 256, 0, stream>>>(delta, np);
    delta_norm_final<<<1, NORM_BLOCKS, 0, stream>>>(np, dnorm);
    // bandwidth-bound LUT apply: 4 px/thread, b128 NT streaming, RT gathers
    trilinear_apply<<<8100, 256, 0, stream>>>(img_full, Lmat, out);
}